// MOERouter_12773232738989
// MI455X (gfx1250) — compile-verified
//
#include <hip/hip_runtime.h>
#include <hip/hip_bf16.h>

typedef __attribute__((ext_vector_type(2))) float v2f;
typedef __attribute__((ext_vector_type(8))) float v8f;
typedef int gv4i __attribute__((vector_size(16)));   // matches builtin param type

constexpr int T    = 16384;   // tokens
constexpr int H    = 2048;    // hidden dim
constexpr int E    = 64;      // experts
constexpr int TOPK = 8;

constexpr int BT  = 64;       // tokens per block
constexpr int KC  = 32;       // K chunk staged per iteration
constexpr int LDX = 36;       // padded LDS row stride (floats): 16B-aligned rows,
                              // gcd(36,64)=4 -> conflict-free half-wave fragment loads

// --------------------------------------------------------------------------
// gfx1250 async global->LDS staging (ASYNCcnt-tracked), with fallback.
// --------------------------------------------------------------------------
#if defined(__has_builtin)
#if __has_builtin(__builtin_amdgcn_global_load_async_to_lds_b128)
#define USE_ASYNC_LDS 1
#endif
#endif
#ifndef USE_ASYNC_LDS
#define USE_ASYNC_LDS 0
#endif

__device__ __forceinline__ void async_wait0()
{
#if defined(__has_builtin) && __has_builtin(__builtin_amdgcn_s_wait_asynccnt)
  __builtin_amdgcn_s_wait_asynccnt(0);
#else
  asm volatile("s_wait_asynccnt 0" ::: "memory");
#endif
}

__device__ __forceinline__ void stage_b128(const float* __restrict__ gsrc,
                                           float* ldst)
{
#if USE_ASYNC_LDS
  __builtin_amdgcn_global_load_async_to_lds_b128(
      (__attribute__((address_space(1))) gv4i*)gsrc,
      (__attribute__((address_space(3))) gv4i*)ldst, 0, 0);
#else
  const float4 v = *reinterpret_cast<const float4*>(gsrc);
  *reinterpret_cast<float4*>(ldst) = v;
#endif
}

// ---------------------------------------------------------------------------
// Kernel 1: router logits GEMM  logits[T,E] = X[T,H] @ W[E,H]^T + b[E]
// fp32 WMMA 16x16x4, double-buffered LDS tiles, 8 waves/block (wave32).
// ---------------------------------------------------------------------------
__global__ __launch_bounds__(256) void moe_router_gemm(
    const float* __restrict__ X, const float* __restrict__ Wg,
    const float* __restrict__ bg, float* __restrict__ logits)
{
  __shared__ float Xs[2][BT * LDX];   // 2 x 9216 B
  __shared__ float Ws[2][E  * LDX];   // 2 x 9216 B

  const int tid  = threadIdx.x;
  const int wave = tid >> 5;
  const int lane = tid & 31;
  const int t0   = blockIdx.x * BT;

  const int tt   = wave & 3;          // token tile (16 tokens) within block
  const int eh   = wave >> 2;         // expert half: 0 -> experts 0..31, 1 -> 32..63
  const int m    = lane & 15;         // M (A) / N (B) index within tile
  const int koff = (lane >> 4) << 1;  // K sub-offset per ISA fragment layout

  v8f acc0 = {};
  v8f acc1 = {};

  const int srow = tid >> 3;          // 0..31 : staging row
  const int scol = (tid & 7) << 2;    // 0,4,...,28 : staging col (float4)

  // Prologue: stage chunk 0 into buffer 0.
#pragma unroll
  for (int p = 0; p < 2; ++p) {
    const int row = p * 32 + srow;    // 0..63
    stage_b128(&X[(size_t)(t0 + row) * H + scol], &Xs[0][row * LDX + scol]);
    stage_b128(&Wg[(size_t)row * H + scol],       &Ws[0][row * LDX + scol]);
  }

  int buf = 0;
  for (int kc = 0; kc < H; kc += KC) {
    // Make buffer `buf` visible to all waves (own asyncs done, then barrier).
    async_wait0();
    __syncthreads();

    // Kick off staging of the next chunk into the other buffer; it overlaps
    // with the WMMA work below and is consumed after next iteration's barrier.
    if (kc + KC < H) {
      const int nxt = buf ^ 1;
      const int kn  = kc + KC;
#pragma unroll
      for (int p = 0; p < 2; ++p) {
        const int row = p * 32 + srow;
        stage_b128(&X[(size_t)(t0 + row) * H + kn + scol],
                   &Xs[nxt][row * LDX + scol]);
        stage_b128(&Wg[(size_t)row * H + kn + scol],
                   &Ws[nxt][row * LDX + scol]);
      }
    }

#pragma unroll
    for (int kk = 0; kk < KC; kk += 4) {
      // A fragment: lane L holds X[t0+tt*16+m][kk+koff .. +1]  (ds_load_b64)
      const v2f a = *reinterpret_cast<const v2f*>(
          &Xs[buf][(tt * 16 + m) * LDX + kk + koff]);
      // B fragments: B[k][n] = W[n][k]
      const v2f b0 = *reinterpret_cast<const v2f*>(
          &Ws[buf][(eh * 32 + m) * LDX + kk + koff]);
      const v2f b1 = *reinterpret_cast<const v2f*>(
          &Ws[buf][(eh * 32 + 16 + m) * LDX + kk + koff]);
      acc0 = __builtin_amdgcn_wmma_f32_16x16x4_f32(
          false, a, false, b0, (short)0, acc0, false, false);
      acc1 = __builtin_amdgcn_wmma_f32_16x16x4_f32(
          false, a, false, b1, (short)0, acc1, false, false);
    }
    buf ^= 1;
  }

  // Epilogue: C layout -> lane holds rows r (lanes 0-15) / 8+r (lanes 16-31)
  const int e0 = eh * 32 + m;
  const int e1 = e0 + 16;
  const float bias0 = bg[e0];
  const float bias1 = bg[e1];
  const int tbase = t0 + tt * 16 + ((lane >> 4) << 3);
#pragma unroll
  for (int r = 0; r < 8; ++r) {
    const size_t off = (size_t)(tbase + r) * E;
    logits[off + e0] = acc0[r] + bias0;
    logits[off + e1] = acc1[r] + bias1;
  }
}

// ---------------------------------------------------------------------------
// Kernel 2: zero the one-hot expert_mask region (d_out is poisoned to 0xAA)
// ---------------------------------------------------------------------------
__global__ __launch_bounds__(256) void zero_mask(int4* __restrict__ p, int n4)
{
  const int i = blockIdx.x * 256 + threadIdx.x;
  if (i < n4) p[i] = make_int4(0, 0, 0, 0);
}

// ---------------------------------------------------------------------------
// Kernel 3: softmax max + top-8 + renormalized weights + one-hot scatter.
// One thread per token. Strict '>' argmax keeps lowest index on ties,
// matching jax.lax.top_k. Softmax denominator cancels in renormalization.
// ---------------------------------------------------------------------------
__global__ __launch_bounds__(256) void router_topk(
    const float* __restrict__ logits, float* __restrict__ weights,
    int* __restrict__ indices, int* __restrict__ mask)
{
  const int t = blockIdx.x * 256 + threadIdx.x;

  float l[E];
  const float4* lp = reinterpret_cast<const float4*>(&logits[(size_t)t * E]);
#pragma unroll
  for (int i = 0; i < E / 4; ++i) {
    const float4 v = lp[i];
    l[4 * i + 0] = v.x; l[4 * i + 1] = v.y;
    l[4 * i + 2] = v.z; l[4 * i + 3] = v.w;
  }

  float mx = l[0];
#pragma unroll
  for (int e = 1; e < E; ++e) mx = fmaxf(mx, l[e]);

  unsigned long long sel = 0ull;
  float wv[TOPK];
  int   iv[TOPK];
  float wsum = 0.0f;
#pragma unroll
  for (int k = 0; k < TOPK; ++k) {
    float best = -3.402823466e38f;
    int bi = 0;
#pragma unroll
    for (int e = 0; e < E; ++e) {
      const bool taken = (sel >> e) & 1ull;
      if (!taken && l[e] > best) { best = l[e]; bi = e; }
    }
    sel |= (1ull << bi);
    const float p = __expf(best - mx);
    wv[k] = p; iv[k] = bi; wsum += p;
  }

  const float inv = 1.0f / wsum;
#pragma unroll
  for (int k = 0; k < TOPK; ++k) {
    weights[(size_t)t * TOPK + k] = wv[k] * inv;
    indices[(size_t)t * TOPK + k] = iv[k];
    // expert_mask[e][k][t] = 1
    mask[(size_t)iv[k] * (TOPK * T) + (size_t)k * T + t] = 1;
  }
}

// ---------------------------------------------------------------------------
extern "C" void kernel_launch(void* const* d_in, const int* in_sizes, int n_in,
                              void* d_out, int out_size, void* d_ws, size_t ws_size,
                              hipStream_t stream)
{
  const float* X  = (const float*)d_in[0];   // [T, H]
  const float* Wg = (const float*)d_in[1];   // [E, H]
  const float* bg = (const float*)d_in[2];   // [E]

  float* logits  = (float*)d_out;                        // [T, E]
  float* weights = logits + (size_t)T * E;               // [T, 8]
  int*   indices = (int*)(weights + (size_t)T * TOPK);   // [T, 8] int32
  int*   mask    = indices + (size_t)T * TOPK;           // [E, 8, T] int32

  moe_router_gemm<<<T / BT, 256, 0, stream>>>(X, Wg, bg, logits);

  const int n4 = (E * TOPK * T) / 4;
  zero_mask<<<(n4 + 255) / 256, 256, 0, stream>>>((int4*)mask, n4);

  router_topk<<<T / 256, 256, 0, stream>>>(logits, weights, indices, mask);

  (void)in_sizes; (void)n_in; (void)out_size; (void)d_ws; (void)ws_size;
}